// NSLoss_33844342292718
// MI455X (gfx1250) — compile-verified
//
#include <hip/hip_runtime.h>
#include <math.h>

#define KNN 5
static constexpr int   B_        = 4;
static constexpr float CONT_W    = 0.5f;
static constexpr float MOM_W     = 0.5f;
static constexpr float DENSITY   = 1.0f;
static constexpr float VISCOSITY = 1.0f;

typedef __attribute__((ext_vector_type(2))) float v2f;
typedef __attribute__((ext_vector_type(8))) float v8f;

// ---------------------------------------------------------------------------
// Kernel 1: brute-force KNN via WMMA Gram tiles.
// One wave (32 threads) per 16-row tile of one batch. d2 = |p|^2+|q|^2-2 p.q,
// with the -2 p.q term computed by v_wmma_f32_16x16x4_f32 (K padded 3->4) and
// the |p|^2+|q|^2 term pre-loaded into the C accumulator, so the WMMA result
// IS the squared-distance tile.
// ---------------------------------------------------------------------------
__global__ __launch_bounds__(32)
void knn_wmma_kernel(const float* __restrict__ xyz, int N, int* __restrict__ nbr)
{
    const int lane    = threadIdx.x;     // 0..31 (wave32)
    const int half    = lane >> 4;       // 0: K={0,1}, 1: K={2,3}
    const int c       = lane & 15;
    const int rowBase = blockIdx.x * 16;
    const int b       = blockIdx.y;
    const float* base = xyz + (size_t)b * N * 3;

    // A operand (16x4, MxK), rows of this tile, scaled by -2.
    const float* pr = base + (size_t)(rowBase + c) * 3;
    float rx = pr[0], ry = pr[1], rz = pr[2];
    float sqr = rx * rx + ry * ry + rz * rz;
    v2f a;
    if (half == 0) { a[0] = -2.0f * rx; a[1] = -2.0f * ry; }
    else           { a[0] = -2.0f * rz; a[1] = 0.0f;       }

    // sqRow[r] = |row point (r + 8*half)|^2  -- matches the C-layout M mapping.
    float sqRow[8];
    #pragma unroll
    for (int r = 0; r < 8; ++r)
        sqRow[r] = __shfl(sqr, r + 8 * half, 32);

    // Per-lane top-5 candidate lists (8 rows x 5), ascending d2.
    float bd[8][KNN];
    int   bi[8][KNN];
    #pragma unroll
    for (int r = 0; r < 8; ++r)
        #pragma unroll
        for (int k = 0; k < KNN; ++k) { bd[r][k] = 3.0e38f; bi[r][k] = 0; }

    const int ntiles = N / 16;
    for (int t = 0; t < ntiles; ++t) {
        const int col   = t * 16 + c;
        const float* pq = base + (size_t)col * 3;
        if (t + 1 < ntiles) __builtin_prefetch(pq + 48, 0, 1);  // next column tile
        float qx = pq[0], qy = pq[1], qz = pq[2];
        float sqc = qx * qx + qy * qy + qz * qz;

        v2f bm;
        if (half == 0) { bm[0] = qx; bm[1] = qy; }
        else           { bm[0] = qz; bm[1] = 0.0f; }

        v8f cm;
        #pragma unroll
        for (int r = 0; r < 8; ++r) cm[r] = sqRow[r] + sqc;

        // d2 tile = (-2A)*B + (|p|^2 + |q|^2)
        cm = __builtin_amdgcn_wmma_f32_16x16x4_f32(
                false, a, false, bm, (short)0, cm, false, false);

        #pragma unroll
        for (int r = 0; r < 8; ++r) {
            const int rowIdx = rowBase + r + 8 * half;
            float d = cm[r];
            if (col != rowIdx && d < bd[r][KNN - 1]) {     // exclude self
                bd[r][KNN - 1] = d; bi[r][KNN - 1] = col;
                #pragma unroll
                for (int k = KNN - 2; k >= 0; --k) {
                    if (bd[r][k + 1] < bd[r][k]) {
                        float td = bd[r][k]; bd[r][k] = bd[r][k + 1]; bd[r][k + 1] = td;
                        int   ti = bi[r][k]; bi[r][k] = bi[r][k + 1]; bi[r][k + 1] = ti;
                    }
                }
            }
        }
    }

    // Merge the 16 per-lane lists of each row through LDS.
    __shared__ float ld_d[16][16][KNN];
    __shared__ int   ld_i[16][16][KNN];
    #pragma unroll
    for (int r = 0; r < 8; ++r) {
        const int row = r + 8 * half;
        #pragma unroll
        for (int k = 0; k < KNN; ++k) {
            ld_d[row][c][k] = bd[r][k];
            ld_i[row][c][k] = bi[r][k];
        }
    }
    __syncthreads();

    if (lane < 16) {
        float* dp = &ld_d[lane][0][0];
        int*   ip = &ld_i[lane][0][0];
        const int outBase = (b * N + rowBase + lane) * KNN;
        for (int s = 0; s < KNN; ++s) {
            float best = 3.0e38f; int bp = 0;
            for (int u = 0; u < 16 * KNN; ++u)
                if (dp[u] < best) { best = dp[u]; bp = u; }
            nbr[outBase + s] = ip[bp];
            dp[bp] = 3.0e38f;
        }
    }
}

// ---------------------------------------------------------------------------
// Kernel 2: per-point continuity term + 3x3 jacobian (stored to workspace).
// ---------------------------------------------------------------------------
__global__ void jac_cont_kernel(const float* __restrict__ xyz,
                                const float* __restrict__ vel,
                                const int*   __restrict__ nbr,
                                float*       __restrict__ jac,
                                float*       __restrict__ cont_acc,
                                int N, int total)
{
    int i = blockIdx.x * blockDim.x + threadIdx.x;
    if (i >= total) return;
    int b = i / N;
    int n = i - b * N;
    const float* xb = xyz + (size_t)b * N * 3;
    const float* vb = vel + (size_t)b * N * 3;

    float pix = xb[n*3+0], piy = xb[n*3+1], piz = xb[n*3+2];
    float vix = vb[n*3+0], viy = vb[n*3+1], viz = vb[n*3+2];

    float c0 = 0.f, c1 = 0.f, c2 = 0.f;
    float J[9] = {0.f,0.f,0.f,0.f,0.f,0.f,0.f,0.f,0.f};

    #pragma unroll
    for (int k = 0; k < KNN; ++k) {
        int j = nbr[i * KNN + k];
        float dx  = pix - xb[j*3+0], dy  = piy - xb[j*3+1], dz  = piz - xb[j*3+2];
        float dvx = vix - vb[j*3+0], dvy = viy - vb[j*3+1], dvz = viz - vb[j*3+2];
        float n2   = dx*dx + dy*dy + dz*dz;
        float invn = rsqrtf(n2);
        float inv2 = 1.0f / n2;
        c0 += dx * dvx * inv2;  c1 += dy * dvy * inv2;  c2 += dz * dvz * inv2;
        float sx = dvx * invn, sy = dvy * invn, sz = dvz * invn;
        J[0] += sx*dx; J[1] += sx*dy; J[2] += sx*dz;
        J[3] += sy*dx; J[4] += sy*dy; J[5] += sy*dz;
        J[6] += sz*dx; J[7] += sz*dy; J[8] += sz*dz;
    }
    const float invK = 1.0f / (float)KNN;
    #pragma unroll
    for (int q = 0; q < 9; ++q) { J[q] *= invK; jac[(size_t)i*9 + q] = J[q]; }

    float cl = fabsf(c0*invK) + fabsf(c1*invK) + fabsf(c2*invK);
    #pragma unroll
    for (int off = 16; off > 0; off >>= 1) cl += __shfl_down(cl, off, 32);
    if ((threadIdx.x & 31) == 0) atomicAdd(cont_acc, cl);
}

// ---------------------------------------------------------------------------
// Kernel 3: laplacian via jacobian differences + momentum term norm.
// ---------------------------------------------------------------------------
__global__ void momentum_kernel(const float* __restrict__ xyz,
                                const float* __restrict__ vel,
                                const int*   __restrict__ nbr,
                                const float* __restrict__ jac,
                                float*       __restrict__ mom_acc,
                                int N, int total)
{
    int i = blockIdx.x * blockDim.x + threadIdx.x;
    if (i >= total) return;
    int b = i / N;
    int n = i - b * N;
    const float* xb = xyz + (size_t)b * N * 3;
    const float* vb = vel + (size_t)b * N * 3;

    float pix = xb[n*3+0], piy = xb[n*3+1], piz = xb[n*3+2];
    float vix = vb[n*3+0], viy = vb[n*3+1], viz = vb[n*3+2];

    float Ji[9];
    #pragma unroll
    for (int q = 0; q < 9; ++q) Ji[q] = jac[(size_t)i*9 + q];

    float l0 = 0.f, l1 = 0.f, l2 = 0.f;
    #pragma unroll
    for (int k = 0; k < KNN; ++k) {
        int j = nbr[i * KNN + k];
        float dx = pix - xb[j*3+0], dy = piy - xb[j*3+1], dz = piz - xb[j*3+2];
        float invn = rsqrtf(dx*dx + dy*dy + dz*dz);
        const float* Jj = jac + ((size_t)b * N + j) * 9;
        float D0 = Ji[0]-Jj[0], D1 = Ji[1]-Jj[1], D2 = Ji[2]-Jj[2];
        float D3 = Ji[3]-Jj[3], D4 = Ji[4]-Jj[4], D5 = Ji[5]-Jj[5];
        float D6 = Ji[6]-Jj[6], D7 = Ji[7]-Jj[7], D8 = Ji[8]-Jj[8];
        l0 += (D0*dx + D1*dy + D2*dz) * invn;
        l1 += (D3*dx + D4*dy + D5*dz) * invn;
        l2 += (D6*dx + D7*dy + D8*dz) * invn;
    }
    const float invK = 1.0f / (float)KNN;
    l0 *= invK; l1 *= invK; l2 *= invK;

    float m0 = DENSITY * (Ji[0]*vix + Ji[1]*viy + Ji[2]*viz) - VISCOSITY * l0;
    float m1 = DENSITY * (Ji[3]*vix + Ji[4]*viy + Ji[5]*viz) - VISCOSITY * l1;
    float m2 = DENSITY * (Ji[6]*vix + Ji[7]*viy + Ji[8]*viz) - VISCOSITY * l2;
    float val = sqrtf(m0*m0 + m1*m1 + m2*m2);

    #pragma unroll
    for (int off = 16; off > 0; off >>= 1) val += __shfl_down(val, off, 32);
    if ((threadIdx.x & 31) == 0) atomicAdd(mom_acc, val);
}

__global__ void init_acc_kernel(float* acc)
{
    if (threadIdx.x < 2) acc[threadIdx.x] = 0.0f;
}

__global__ void finalize_kernel(const float* __restrict__ acc,
                                float* __restrict__ out, int total)
{
    float cont_mean = acc[0] / (float)(total * 3);
    float mom_mean  = acc[1] / (float)total;
    out[0] = CONT_W * cont_mean + MOM_W * mom_mean;
}

// ---------------------------------------------------------------------------
extern "C" void kernel_launch(void* const* d_in, const int* in_sizes, int n_in,
                              void* d_out, int out_size, void* d_ws, size_t ws_size,
                              hipStream_t stream)
{
    const float* xyz = (const float*)d_in[0];   // target_tensor: positions
    const float* vel = (const float*)d_in[1];   // pred_tensor: velocities
    float* out = (float*)d_out;

    const int total = in_sizes[0] / 3;          // B*N = 32768
    const int N     = total / B_;               // 8192

    // Workspace layout: [0..255] accumulators, then nbr idx, then jacobians.
    char*  ws  = (char*)d_ws;
    float* acc = (float*)ws;                                         // 2 floats
    int*   nbr = (int*)(ws + 256);                                   // total*5 ints
    float* jac = (float*)(ws + 256 + (size_t)total * KNN * sizeof(int)); // total*9 f32

    init_acc_kernel<<<1, 32, 0, stream>>>(acc);

    dim3 gknn(N / 16, B_);
    knn_wmma_kernel<<<gknn, 32, 0, stream>>>(xyz, N, nbr);

    const int threads = 256;
    const int blocks  = (total + threads - 1) / threads;
    jac_cont_kernel<<<blocks, threads, 0, stream>>>(xyz, vel, nbr, jac, acc + 0, N, total);
    momentum_kernel<<<blocks, threads, 0, stream>>>(xyz, vel, nbr, jac, acc + 1, N, total);

    finalize_kernel<<<1, 1, 0, stream>>>(acc, out, total);
}